// FunctPolynomial_80659485819633
// MI455X (gfx1250) — compile-verified
//
#include <hip/hip_runtime.h>
#include <cstdint>

typedef float v2f __attribute__((ext_vector_type(2)));
typedef float v8f __attribute__((ext_vector_type(8)));
typedef unsigned int v4u __attribute__((ext_vector_type(4)));
typedef int v8i __attribute__((ext_vector_type(8)));
typedef int v4i __attribute__((ext_vector_type(4)));

#define LL 1024
#define NBATCH 64
#define HROWS 32
#define TPB 256
#define TOTAL_ROWS (NBATCH * LL) /* 65536 rows when viewing x as 2D [rows x 1024] */

// ---------------------------------------------------------------------------
// TDM: DMA `nrows` consecutive rows of 1024 fp32 from global into LDS.
// D# built per CDNA5 ISA §8 (group0 128b, group1 256b; groups 2/3 zero => 2D).
// 6-arg builtin form (clang-23 / therock toolchain).
// ---------------------------------------------------------------------------
__device__ __forceinline__ void tdm_load_rows(const float* gptr_row0,
                                              unsigned lds_byte, int nrows) {
  unsigned long long ga = (unsigned long long)(uintptr_t)gptr_row0;
  v4u g0;
  g0[0] = 1u;                                           // count=1 (valid), user desc
  g0[1] = lds_byte;                                     // lds_addr
  g0[2] = (unsigned)(ga & 0xFFFFFFFFull);               // global_addr[31:0]
  g0[3] = (unsigned)((ga >> 32) & 0x01FFFFFFull)        // global_addr[56:32]
          | 0x80000000u;                                // type=2 ("image")
  v8i g1;
  g1[0] = (int)(2u << 16);                              // data_size=4B; no iterate/pad
  g1[1] = (int)((LL & 0xFFFF) << 16);                   // tensor_dim0[15:0]
  g1[2] = (int)(((LL >> 16) & 0xFFFF) |
                ((TOTAL_ROWS & 0xFFFF) << 16));         // dim0 hi | tensor_dim1 lo
  g1[3] = (int)(((TOTAL_ROWS >> 16) & 0xFFFF) |
                ((LL & 0xFFFF) << 16));                 // dim1 hi | tile_dim0=1024
  g1[4] = nrows & 0xFFFF;                               // tile_dim1 (tile_dim2=0 => 2D)
  g1[5] = LL;                                           // tensor_dim0_stride[31:0]
  g1[6] = 0;                                            // stride0 hi | stride1 lo
  g1[7] = 0;
  v4i gz4 = (v4i){0, 0, 0, 0};
  v8i gz8 = (v8i){0, 0, 0, 0, 0, 0, 0, 0};
  __builtin_amdgcn_tensor_load_to_lds(g0, g1, gz4, gz4, gz8, 0);
}

__global__ void zero_ws_kernel(float* p) {
  p[threadIdx.x] = 0.0f; // 512 floats (64 batches x 8 slots)
}

// ---------------------------------------------------------------------------
// Main: one block = one 32-row strip of one image. TDM -> LDS, stencil sums,
// wave32 butterfly, WMMA cross-wave reduction, global fp32 atomics.
// ---------------------------------------------------------------------------
__global__ __launch_bounds__(TPB) void lattice_kernel(const float* __restrict__ x,
                                                      float* __restrict__ partial) {
  extern __shared__ float smem[];                 // (H+2)*1024 tile + red[8*6]
  float* tile = smem;
  float* red  = smem + (HROWS + 2) * LL;

  const int strip = blockIdx.x & (LL / HROWS - 1);   // 32 strips / image
  const int batch = blockIdx.x >> 5;
  const int r0    = strip * HROWS;

  // --- stage 1: TDM loads (wave 0 only; TENSORcnt is per-wave) --------------
  if (threadIdx.x < 32) {
    const unsigned lds_base = (unsigned)(uintptr_t)tile;
    const size_t rtop = (size_t)batch * LL + ((r0 + LL - 1) & (LL - 1));
    const size_t rbod = (size_t)batch * LL + r0;
    const size_t rbot = (size_t)batch * LL + ((r0 + HROWS) & (LL - 1));
    tdm_load_rows(x + rtop * LL, lds_base, 1);                       // halo top
    tdm_load_rows(x + rbod * LL, lds_base + 1 * LL * 4, HROWS);      // body
    tdm_load_rows(x + rbot * LL, lds_base + (HROWS + 1) * LL * 4, 1);// halo bottom
    __builtin_amdgcn_s_wait_tensorcnt(0);
  }
  __syncthreads();

  // --- stage 2: stencil + 6 partial sums -----------------------------------
  const int tid = threadIdx.x;
  const int j   = tid * 4;                        // this thread's 4 columns
  const int jm1 = (j == 0) ? (LL - 1) : (j - 1);
  const int jp4 = (j + 4) & (LL - 1);

  float s1 = 0.f, s3 = 0.f, s4 = 0.f, s7 = 0.f, sQ = 0.f, s8 = 0.f;

#pragma unroll 4
  for (int k = 0; k < HROWS; ++k) {
    const float* rowC = tile + (k + 1) * LL;
    const float* rowU = tile + k * LL;
    const float* rowD = tile + (k + 2) * LL;
    const float4 phi = *(const float4*)(rowC + j);   // ds_load_b128
    const float4 up  = *(const float4*)(rowU + j);
    const float4 dn  = *(const float4*)(rowD + j);
    const float right = rowC[jp4];
    const float left  = rowC[jm1];

    const float ph[4] = {phi.x, phi.y, phi.z, phi.w};
    const float ta[4] = {phi.y, phi.z, phi.w, right};   // phi_tau = right neighbor
    const float lf[4] = {left, phi.x, phi.y, phi.z};
    const float uu[4] = {up.x, up.y, up.z, up.w};
    const float dd[4] = {dn.x, dn.y, dn.z, dn.w};
#pragma unroll
    for (int e = 0; e < 4; ++e) {
      const float nb = (uu[e] + dd[e]) + (lf[e] + ta[e]);
      const float p  = ph[e] * ta[e];                // phi*phi_tau
      const float q  = ph[e] * ph[e];                // phi^2
      s1 += p;
      s3 += q;
      s4 = fmaf(p, p, s4);                           // (phi*phi_tau)^2
      s7 = fmaf(ph[e], nb, s7);                      // phi*neighbors
      sQ = fmaf(ta[e], nb, sQ);                      // phi_tau*neighbors
      s8 = fmaf(p, q, s8);                           // phi^3*phi_tau
    }
  }

  // --- stage 3: wave32 butterfly reduction ---------------------------------
#pragma unroll
  for (int m = 16; m >= 1; m >>= 1) {
    s1 += __shfl_xor(s1, m, 32);
    s3 += __shfl_xor(s3, m, 32);
    s4 += __shfl_xor(s4, m, 32);
    s7 += __shfl_xor(s7, m, 32);
    sQ += __shfl_xor(sQ, m, 32);
    s8 += __shfl_xor(s8, m, 32);
  }
  const int lane = tid & 31, wave = tid >> 5;
  if (lane == 0) {
    float* w = red + wave * 6;
    w[0] = s1; w[1] = s3; w[2] = s4; w[3] = s7; w[4] = sQ; w[5] = s8;
  }
  __syncthreads();

  // --- stage 4: cross-wave reduce with v_wmma_f32_16x16x4_f32 --------------
  // A(16x4) holds red[wave][comp] in the documented A layout; B = ones (layout
  // invariant), so D[m][n] = sum_k A[m][k]. Two accumulating WMMAs cover the
  // 8 waves; lane 0's D VGPRs 0..5 are the 6 block-level sums.
  if (threadIdx.x < 32) {               // EXEC all-ones within wave 0
    const int m = lane & 15;
    const int hi = (lane >= 16) ? 1 : 0;
    const bool valid = (m < 6);
    v2f a0, a1, bones;
    a0.x = valid ? red[(hi ? 2 : 0) * 6 + m] : 0.f;   // A[m][0] / A[m][2]
    a0.y = valid ? red[(hi ? 3 : 1) * 6 + m] : 0.f;   // A[m][1] / A[m][3]
    a1.x = valid ? red[(hi ? 6 : 4) * 6 + m] : 0.f;
    a1.y = valid ? red[(hi ? 7 : 5) * 6 + m] : 0.f;
    bones.x = 1.0f; bones.y = 1.0f;
    v8f c = {};
    c = __builtin_amdgcn_wmma_f32_16x16x4_f32(false, a0, false, bones,
                                              (short)0, c, false, false);
    c = __builtin_amdgcn_wmma_f32_16x16x4_f32(false, a1, false, bones,
                                              (short)0, c, false, false);
    if (lane == 0) {
#pragma unroll
      for (int p = 0; p < 6; ++p)
        atomicAdd(&partial[batch * 8 + p], c[p]);     // global_atomic_add_f32
    }
  }
}

// ---------------------------------------------------------------------------
// Finalize: P5 = P7 - 4*P3, P6 = Q - 4*P1, P2 = P1^2; dot with coeffs.
// ---------------------------------------------------------------------------
__global__ void finalize_kernel(const float* __restrict__ partial,
                                const float* __restrict__ coeffs,
                                float* __restrict__ out) {
  const int b = threadIdx.x;
  if (b >= NBATCH) return;
  const float inv = 1.0f / ((float)LL * (float)LL);
  const float s1 = partial[b * 8 + 0], s3 = partial[b * 8 + 1];
  const float s4 = partial[b * 8 + 2], s7 = partial[b * 8 + 3];
  const float sQ = partial[b * 8 + 4], s8 = partial[b * 8 + 5];
  const float P1 = s1 * inv;
  const float P2 = P1 * P1;
  const float P3 = s3 * inv;
  const float P4 = s4 * inv;
  const float P5 = (s7 - 4.0f * s3) * inv;
  const float P6 = (sQ - 4.0f * s1) * inv;
  const float P7 = s7 * inv;
  const float P8 = s8 * inv;
  out[b] = coeffs[0] * P1 + coeffs[1] * P2 + coeffs[2] * P3 + coeffs[3] * P4 +
           coeffs[4] * P5 + coeffs[5] * P6 + coeffs[6] * P7 + coeffs[7] * P8;
}

extern "C" void kernel_launch(void* const* d_in, const int* in_sizes, int n_in,
                              void* d_out, int out_size, void* d_ws, size_t ws_size,
                              hipStream_t stream) {
  (void)in_sizes; (void)n_in; (void)out_size; (void)ws_size;
  const float* x      = (const float*)d_in[0];
  const float* coeffs = (const float*)d_in[1];
  float* out     = (float*)d_out;
  float* partial = (float*)d_ws;                   // 64 batches x 8 floats

  zero_ws_kernel<<<1, 512, 0, stream>>>(partial);

  const size_t shmem = ((size_t)(HROWS + 2) * LL + 64) * sizeof(float);
  lattice_kernel<<<dim3(NBATCH * (LL / HROWS)), dim3(TPB), shmem, stream>>>(x, partial);

  finalize_kernel<<<1, 64, 0, stream>>>(partial, coeffs, out);
}